// BigBirdAttention_2740189134921
// MI455X (gfx1250) — compile-verified
//
#include <hip/hip_runtime.h>
#include <math.h>

// BigBird sparse attention fwd, MI455X (gfx1250), f32 WMMA 16x16x4.
// B=2 H=16 M=4096 D=64, 64x64 blocks, R=3 random blocks.

#define BB   2
#define HH   16
#define MM   4096
#define DD   64
#define NBLK 64
#define RRND 3
#define NEGV (-10000.0f)
#define SCALE 0.125f   // 1/sqrt(64)

typedef float v2f __attribute__((ext_vector_type(2)));
typedef float v8f __attribute__((ext_vector_type(8)));

__device__ __forceinline__ v8f wmma_f32(v2f a, v2f b, v8f c) {
  // (neg_a, A, neg_b, B, c_mod, C, reuse_a, reuse_b) -> v_wmma_f32_16x16x4_f32
  return __builtin_amdgcn_wmma_f32_16x16x4_f32(false, a, false, b, (short)0, c,
                                               false, false);
}

// ---- LDS layout (float offsets) ----
#define QSTR   68
#define OFF_Q  0                         // 64*68   = 4352
#define SCSTR  516
#define OFF_SC 4352                      // 64*516  = 33024
#define OFF_TM 37376                     // 512 key to-masks (per chunk)
#define OFF_FQ 37888                     // 64 query from-masks (== seq_m)
#define OFF_RM 37952                     // 64 row maxes
#define OFF_RS 38016                     // 64 row sums
#define OFF_PB 38080                     // 256 partial reduction buffer
#define SMEM_FLOATS 38336                // 153,344 bytes  (<< 320KB LDS/WGP)

// Gathered key-block id for slot j of chunk c.
__device__ __forceinline__ int blk_of(int qb, int full, int c, int j,
                                      int r0, int r1, int r2) {
  if (full) return c * 8 + j;                             // dense rows, natural order
  int rr = (j == 4) ? r0 : (j == 5) ? r1 : r2;
  if (qb == 1)
    return (j == 0) ? 0 : (j == 1) ? 1 : (j == 2) ? 2 : (j == 3) ? 63 : rr;
  if (qb == NBLK - 2)
    return (j == 0) ? 0 : (j <= 3) ? (60 + j) : rr;       // 0,61,62,63,r*
  // middle: first, qb-1, qb, qb+1, r0..r2, last
  return (j == 0) ? 0 : (j <= 3) ? (qb - 2 + j) : (j <= 6) ? rr : 63;
}
// Mask mode: 0 -> (1 - to[k])*NEG ; 1 -> (1 - from[q]*to[k])*NEG
__device__ __forceinline__ int mode_of(int qb, int full, int j) {
  if (full) return 0;
  if (qb == 1 || qb == NBLK - 2) return (j < 4) ? 0 : 1;
  return (j == 0 || j == 7) ? 0 : 1;
}

__global__ void __launch_bounds__(256)
bigbird_fwd(const float* __restrict__ Q, const float* __restrict__ K,
            const float* __restrict__ V, const float* __restrict__ msk,
            const int* __restrict__ rnd, float* __restrict__ out_ctx,
            float* __restrict__ out_attn) {
  extern __shared__ float sm[];
  const int qb = blockIdx.x, h = blockIdx.y, b = blockIdx.z;
  const int bh = b * HH + h;
  const int tid = threadIdx.x;
  const int wave = tid >> 5, lane = tid & 31;
  const int hl = lane >> 4, lm = lane & 15;   // half-wave, lane-in-half

  const int full   = (qb == 0 || qb == NBLK - 1);
  const int middle = (qb >= 2 && qb <= NBLK - 3);
  const int nblkc  = (qb == 1 || qb == NBLK - 2) ? 7 : 8;  // blocks per chunk
  const int nk     = nblkc * 64;                           // keys per chunk
  const int C      = full ? 8 : 1;                         // chunks

  int r0 = 0, r1 = 0, r2 = 0;
  if (!full) {
    const int* rp = rnd + ((size_t)bh * (NBLK - 2) + (qb - 1)) * RRND;
    r0 = rp[0]; r1 = rp[1]; r2 = rp[2];
  }

  const size_t baseQK = (size_t)bh * MM * DD;

  // Stage Q block (64x64) in LDS; init masks/stats.
  for (int idx = tid; idx < 64 * DD; idx += 256) {
    int qr = idx >> 6, qc = idx & 63;
    sm[OFF_Q + qr * QSTR + qc] = Q[baseQK + (size_t)(qb * 64 + qr) * DD + qc];
  }
  if (tid < 64) {
    sm[OFF_FQ + tid] = msk[b * MM + qb * 64 + tid];  // from-mask == seq_m row mask
    sm[OFF_RM + tid] = -3.0e38f;
    sm[OFF_RS + tid] = 0.0f;
  }
  __syncthreads();

  v8f accA = {}, accB = {};        // two 16x16 PV output tiles per wave
  const int nkt = nk >> 4;

  for (int pass = 0; pass < 2; ++pass) {       // pass0: row max; pass1: exp/sum + PV
    for (int c = 0; c < C; ++c) {
      const bool compute = (pass == 0) || full;   // gathered path keeps scores in LDS
      if (compute) {
        // chunk to-masks
        for (int k = tid; k < nk; k += 256) {
          int blk = blk_of(qb, full, c, k >> 6, r0, r1, r2);
          sm[OFF_TM + k] = msk[b * MM + blk * 64 + (k & 63)];
        }
        __syncthreads();   // also fences prior-chunk PV reads of sc
        // ---- scores: S = (Q K^T)*scale + mask, via f32 WMMA 16x16x4 ----
        const int ntiles = 4 * nkt;
        for (int t = wave; t < ntiles; t += 8) {
          int qs = t & 3, kt = t >> 2;
          int m0 = qs * 16, kt16 = kt * 16;
          int j = kt16 >> 6;
          int blk = blk_of(qb, full, c, j, r0, r1, r2);
          int md  = mode_of(qb, full, j);
          const float* krow = K + baseQK + (size_t)(blk * 64 + (kt16 & 63) + lm) * DD;
          const float* qrow = &sm[OFF_Q + (m0 + lm) * QSTR];
          v8f acc = {};
          #pragma unroll
          for (int kb = 0; kb < DD; kb += 4) {
            int kd = kb + 2 * hl;
            v2f a;  a.x = qrow[kd];  a.y = qrow[kd + 1];
            v2f bb; bb.x = krow[kd]; bb.y = krow[kd + 1];
            acc = wmma_f32(a, bb, acc);
          }
          float tm = sm[OFF_TM + kt16 + lm];     // this lane's key column mask
          #pragma unroll
          for (int r = 0; r < 8; ++r) {
            int m = m0 + r + 8 * hl;
            float mv = md ? (sm[OFF_FQ + m] * tm) : tm;
            sm[OFF_SC + m * SCSTR + kt16 + lm] = acc[r] * SCALE + (1.0f - mv) * NEGV;
          }
        }
      }
      __syncthreads();
      const int row = tid & 63, part = tid >> 6;
      if (pass == 0) {
        // running row max
        float mx = -3.0e38f;
        for (int k = part; k < nk; k += 4)
          mx = fmaxf(mx, sm[OFF_SC + row * SCSTR + k]);
        sm[OFF_PB + part * 64 + row] = mx;
        __syncthreads();
        if (tid < 64) {
          float m4 = fmaxf(fmaxf(sm[OFF_PB + tid], sm[OFF_PB + 64 + tid]),
                           fmaxf(sm[OFF_PB + 128 + tid], sm[OFF_PB + 192 + tid]));
          sm[OFF_RM + tid] = fmaxf(sm[OFF_RM + tid], m4);
        }
        __syncthreads();
      } else {
        // probs + row sums (overwrite scores in LDS with exp(s-max))
        float rm = sm[OFF_RM + row];
        float s = 0.0f;
        for (int k = part; k < nk; k += 4) {
          float p = __expf(sm[OFF_SC + row * SCSTR + k] - rm);
          sm[OFF_SC + row * SCSTR + k] = p;
          s += p;
        }
        sm[OFF_PB + part * 64 + row] = s;
        __syncthreads();
        if (tid < 64)
          sm[OFF_RS + tid] += sm[OFF_PB + tid] + sm[OFF_PB + 64 + tid] +
                              sm[OFF_PB + 128 + tid] + sm[OFF_PB + 192 + tid];
        __syncthreads();
        // ---- PV accumulate: O += P V, f32 WMMA 16x16x4 ----
        #pragma unroll
        for (int tt = 0; tt < 2; ++tt) {
          int t = wave + tt * 8;
          int ms = t >> 2, ns = t & 3;
          int m0 = ms * 16, n0 = ns * 16;
          v8f acc = tt ? accB : accA;
          for (int j = 0; j < nblkc; ++j) {
            int blk = blk_of(qb, full, c, j, r0, r1, r2);
            const float* Vb = V + baseQK + (size_t)(blk * 64) * DD;
            int kbase = j * 64;
            #pragma unroll 4
            for (int kw = 0; kw < 64; kw += 4) {
              int kd = kw + 2 * hl;
              v2f a;  a.x = sm[OFF_SC + (m0 + lm) * SCSTR + kbase + kd];
                      a.y = sm[OFF_SC + (m0 + lm) * SCSTR + kbase + kd + 1];
              v2f bb; bb.x = Vb[(size_t)kd * DD + n0 + lm];
                      bb.y = Vb[(size_t)(kd + 1) * DD + n0 + lm];
              acc = wmma_f32(a, bb, acc);
            }
          }
          if (tt) accB = acc; else accA = acc;
        }
      }
    }
  }
  __syncthreads();

  // attn output (middle blocks only): normalized 64x512 panel, streamed NT.
  if (middle) {
    float* ap = out_attn + ((size_t)bh * (NBLK - 4) + (qb - 2)) * 64 * 512;
    for (int idx = tid; idx < 64 * 512; idx += 256) {
      int q = idx >> 9, k = idx & 511;
      float v = sm[OFF_SC + q * SCSTR + k] / sm[OFF_RS + q];
      __builtin_nontemporal_store(v, ap + (size_t)q * 512 + k);
    }
  }

  // context: normalize, apply seq_m, transpose to (b, m, h, d).
  #pragma unroll
  for (int tt = 0; tt < 2; ++tt) {
    int t = wave + tt * 8;
    int ms = t >> 2, ns = t & 3;
    v8f acc = tt ? accB : accA;
    #pragma unroll
    for (int r = 0; r < 8; ++r) {
      int m = ms * 16 + r + 8 * hl;
      float val = acc[r] / sm[OFF_RS + m] * sm[OFF_FQ + m];
      out_ctx[((size_t)(b * MM + qb * 64 + m) * HH + h) * DD + ns * 16 + lm] = val;
    }
  }
}

extern "C" void kernel_launch(void* const* d_in, const int* in_sizes, int n_in,
                              void* d_out, int out_size, void* d_ws, size_t ws_size,
                              hipStream_t stream) {
  (void)in_sizes; (void)n_in; (void)out_size; (void)d_ws; (void)ws_size;
  const float* Q    = (const float*)d_in[0];
  const float* Km   = (const float*)d_in[1];
  const float* V    = (const float*)d_in[2];
  const float* mask = (const float*)d_in[3];
  const int*   rnd  = (const int*)d_in[4];
  float* out_ctx  = (float*)d_out;
  float* out_attn = out_ctx + (size_t)BB * HH * MM * DD;   // tuple: (context, attn)

  dim3 grid(NBLK, HH, BB);
  size_t smem = (size_t)SMEM_FLOATS * sizeof(float);
  hipLaunchKernelGGL(bigbird_fwd, grid, dim3(256), smem, stream,
                     Q, Km, V, mask, rnd, out_ctx, out_attn);
}